// LinearAttention_71305047048624
// MI455X (gfx1250) — compile-verified
//
#include <hip/hip_runtime.h>
#include <hip/hip_bf16.h>

#define B_ 4
#define T_ 512
#define D_ 192
#define BT_ 2048
#define TD_ (T_ * D_)
#define CHUNK 64
#define NCHUNK 8
#define LBLK 48
#define NLBLK 4

#define OUT_OFF_S 393216   // B*T*D
#define OUT_OFF_Z 540672   // + B*D*D

typedef __attribute__((ext_vector_type(2))) float v2f;
typedef __attribute__((ext_vector_type(8))) float v8f;

__device__ __forceinline__ v8f wmma_f32_k4(v2f a, v2f b, v8f c) {
  // D = A(16x4, f32) * B(4x16, f32) + C(16x16, f32)
  return __builtin_amdgcn_wmma_f32_16x16x4_f32(false, a, false, b, (short)0, c,
                                               false, false);
}

__device__ __forceinline__ float clamp20(float v) {
  return fminf(fmaxf(v, -1e20f), 1e20f);
}

// ---------------------------------------------------------------------------
// Kernel 1: qkv = x @ W^T + b ; Q,K get elu()+1 ; scatter into Q/K/V buffers.
// Grid: 128 blocks (one 16-row tile of BT=2048 each), 1 wave per block.
// A-fragments for the whole 192-deep K loop are register-cached (48 x v2f).
// ---------------------------------------------------------------------------
__global__ __launch_bounds__(32) void qkv_proj_kernel(
    const float* __restrict__ x, const float* __restrict__ W,
    const float* __restrict__ bias, float* __restrict__ Qb,
    float* __restrict__ Kb, float* __restrict__ Vb) {
  const int lane = threadIdx.x & 31;
  const int l15 = lane & 15;
  const int lhalf = lane >> 4;  // 0 or 1
  const int m0 = blockIdx.x * 16;

  // Preload A fragments: A[m][k], m = m0+l15, k = 4s + lhalf*2 + {0,1}
  v2f af[48];
  const float* xrow = x + (size_t)(m0 + l15) * D_ + lhalf * 2;
#pragma unroll
  for (int s = 0; s < 48; ++s) {
    af[s].x = xrow[4 * s];
    af[s].y = xrow[4 * s + 1];
  }

  for (int nt = 0; nt < 36; ++nt) {
    const int n0 = nt * 16;
    // B[k][n] = W[n][k] (W is row-major (576,192))
    const float* wrow = W + (size_t)(n0 + l15) * D_ + lhalf * 2;
    v8f acc = {};
#pragma unroll
    for (int s = 0; s < 48; ++s) {
      v2f bf;
      bf.x = wrow[4 * s];
      bf.y = wrow[4 * s + 1];
      acc = wmma_f32_k4(af[s], bf, acc);
    }
    // Epilogue: bias, elu+1 for Q/K columns, scatter. col range is one
    // 16-wide tile so the 192/384 splits are wave-uniform.
    const int col = n0 + l15;
    const float bv = bias[col];
#pragma unroll
    for (int r = 0; r < 8; ++r) {
      const int row = m0 + r + lhalf * 8;
      float v = acc[r] + bv;
      if (col < 384) v = (v > 0.f) ? (v + 1.f) : __expf(v);  // elu(v)+1
      if (col < 192)
        Qb[(size_t)row * D_ + col] = v;
      else if (col < 384)
        Kb[(size_t)row * D_ + (col - 192)] = v;
      else
        Vb[(size_t)row * D_ + (col - 384)] = v;
    }
  }
}

// ---------------------------------------------------------------------------
// Kernel 2: chunked linear-attention scan.
// Grid: (NLBLK, B). Each block owns state slice S[:, l0:l0+48] (192x48) in LDS
// and walks the 8 chunks of 64 timesteps sequentially.
// ---------------------------------------------------------------------------
#define LDQ 196  // 192 + 4 pad (bank-conflict break)
#define LDV 52   // 48 + 4
#define LDA 68   // 64 + 4
#define OFF_Q 0
#define OFF_K 12544  // 64*196
#define OFF_V 25088  // +64*196
#define OFF_A 28416  // +64*52
#define OFF_S 32768  // +64*68
#define OFF_Z 42752  // +192*52
#define OFF_DEN 42944
#define SMEM_FLOATS 43008

__global__ __launch_bounds__(256) void linattn_scan_kernel(
    const float* __restrict__ Qb, const float* __restrict__ Kb,
    const float* __restrict__ Vb, const float* __restrict__ S0,
    const float* __restrict__ Z0, float* __restrict__ out) {
  extern __shared__ float smem[];
  float* sQ = smem + OFF_Q;     // [64][LDQ]
  float* sK = smem + OFF_K;     // [64][LDQ]
  float* sV = smem + OFF_V;     // [64][LDV]
  float* sA = smem + OFF_A;     // [64][LDA]
  float* sS = smem + OFF_S;     // [192][LDV]
  float* sZ = smem + OFF_Z;     // [192]
  float* sDen = smem + OFF_DEN; // [64]

  const int tid = threadIdx.x;
  const int lane = tid & 31;
  const int wave = tid >> 5;
  const int l15 = lane & 15;
  const int lhalf = lane >> 4;
  const int batch = blockIdx.y;
  const int l0 = blockIdx.x * LBLK;

  const float* Qg = Qb + (size_t)batch * TD_;
  const float* Kg = Kb + (size_t)batch * TD_;
  const float* Vg = Vb + (size_t)batch * TD_;
  float* Og = out + (size_t)batch * TD_;

  // Init state from S0 / Z0 (zeros in practice, but be faithful).
  for (int idx = tid; idx < 192 * LBLK; idx += 256) {
    const int i = idx / LBLK, l = idx % LBLK;
    sS[i * LDV + l] = S0[(size_t)batch * (D_ * D_) + i * D_ + l0 + l];
  }
  for (int i = tid; i < 192; i += 256) sZ[i] = Z0[(size_t)batch * D_ + i];
  __syncthreads();

  for (int c = 0; c < NCHUNK; ++c) {
    const int t0 = c * CHUNK;

    // ---- Load Q,K chunks (64x192) and V slice (64x48) via float4 ----
    for (int idx = tid; idx < 64 * 48; idx += 256) {  // 48 float4 per 192 row
      const int row = idx / 48, cv = idx % 48;
      const float4 q = *(const float4*)(Qg + (size_t)(t0 + row) * D_ + cv * 4);
      const float4 k = *(const float4*)(Kg + (size_t)(t0 + row) * D_ + cv * 4);
      *(float4*)(sQ + row * LDQ + cv * 4) = q;
      *(float4*)(sK + row * LDQ + cv * 4) = k;
    }
    for (int idx = tid; idx < 64 * 12; idx += 256) {  // 12 float4 per 48 slice
      const int row = idx / 12, cv = idx % 12;
      const float4 v =
          *(const float4*)(Vg + (size_t)(t0 + row) * D_ + l0 + cv * 4);
      *(float4*)(sV + row * LDV + cv * 4) = v;
    }
    __syncthreads();

    // ---- Phase A: A = Q K^T  (64x64, inner 192) ----
    for (int tt = wave; tt < 16; tt += 8) {
      const int am0 = (tt >> 2) * 16, an0 = (tt & 3) * 16;
      const float* qp = sQ + (am0 + l15) * LDQ + lhalf * 2;
      const float* kp = sK + (an0 + l15) * LDQ + lhalf * 2;  // B[k][n]=K[n][k]
      v8f acc = {};
#pragma unroll
      for (int s = 0; s < 48; ++s) {
        v2f a, b;
        a.x = qp[4 * s];  a.y = qp[4 * s + 1];
        b.x = kp[4 * s];  b.y = kp[4 * s + 1];
        acc = wmma_f32_k4(a, b, acc);
      }
#pragma unroll
      for (int r = 0; r < 8; ++r)
        sA[(am0 + r + lhalf * 8) * LDA + an0 + l15] = acc[r];
    }
    __syncthreads();

    // ---- Phase B: causal mask (s<=t), row-sum, den_t = rs + Q_t.Z + 1e-5 ----
    if (tid < 64) {
      const int t = tid;
      float rs = 0.f;
      for (int s = 0; s < 64; ++s) {
        const float a = sA[t * LDA + s];
        if (s > t) sA[t * LDA + s] = 0.f;
        else rs += a;
      }
      float qz = 0.f;
      for (int d = 0; d < 192; ++d) qz += sQ[t * LDQ + d] * sZ[d];
      sDen[t] = rs + qz + 1e-5f;
    }
    __syncthreads();

    // ---- Phase C: O = mask(A)@V + Q@S ; divide by den ; store ----
    for (int tt = wave; tt < 12; tt += 8) {
      const int om0 = (tt / 3) * 16, on0 = (tt % 3) * 16;
      v8f acc = {};
      const float* ap = sA + (om0 + l15) * LDA + lhalf * 2;
#pragma unroll
      for (int s = 0; s < 16; ++s) {  // intra-chunk, inner 64
        const int kk = 4 * s + lhalf * 2;
        v2f a, b;
        a.x = ap[4 * s];              a.y = ap[4 * s + 1];
        b.x = sV[kk * LDV + on0 + l15]; b.y = sV[(kk + 1) * LDV + on0 + l15];
        acc = wmma_f32_k4(a, b, acc);
      }
      const float* qp = sQ + (om0 + l15) * LDQ + lhalf * 2;
#pragma unroll
      for (int s = 0; s < 48; ++s) {  // inter-chunk Q @ S_prev, inner 192
        const int kk = 4 * s + lhalf * 2;
        v2f a, b;
        a.x = qp[4 * s];              a.y = qp[4 * s + 1];
        b.x = sS[kk * LDV + on0 + l15]; b.y = sS[(kk + 1) * LDV + on0 + l15];
        acc = wmma_f32_k4(a, b, acc);
      }
#pragma unroll
      for (int r = 0; r < 8; ++r) {
        const int t = om0 + r + lhalf * 8;
        Og[(size_t)(t0 + t) * D_ + l0 + on0 + l15] = acc[r] / sDen[t];
      }
    }
    __syncthreads();

    // ---- Phase D: S += K^T V  (192x48, inner 64) ; Z += colsum(K) ----
    for (int tt = wave; tt < 36; tt += 8) {
      const int sm0 = (tt / 3) * 16, sn0 = (tt % 3) * 16;
      v8f acc;
#pragma unroll
      for (int r = 0; r < 8; ++r)
        acc[r] = sS[(sm0 + r + lhalf * 8) * LDV + sn0 + l15];
#pragma unroll
      for (int s = 0; s < 16; ++s) {
        const int kk = 4 * s + lhalf * 2;
        v2f a, b;  // A[i][kk] = K[kk][i]  (K transposed)
        a.x = sK[kk * LDQ + sm0 + l15]; a.y = sK[(kk + 1) * LDQ + sm0 + l15];
        b.x = sV[kk * LDV + sn0 + l15]; b.y = sV[(kk + 1) * LDV + sn0 + l15];
        acc = wmma_f32_k4(a, b, acc);
      }
#pragma unroll
      for (int r = 0; r < 8; ++r)
        sS[(sm0 + r + lhalf * 8) * LDV + sn0 + l15] = acc[r];
    }
    if (tid < 192) {
      float z = sZ[tid];
      for (int s = 0; s < 64; ++s) z += sK[s * LDQ + tid];
      sZ[tid] = z;
    }
    __syncthreads();
  }

  // ---- Final: S_last (clamped) and Z_last ----
  float* Sout = out + OUT_OFF_S + (size_t)batch * (D_ * D_);
  for (int idx = tid; idx < 192 * LBLK; idx += 256) {
    const int i = idx / LBLK, l = idx % LBLK;
    Sout[i * D_ + l0 + l] = clamp20(sS[i * LDV + l]);
  }
  if (blockIdx.x == 0 && tid < 192)
    out[OUT_OFF_Z + (size_t)batch * D_ + tid] = clamp20(sZ[tid]);
}

// ---------------------------------------------------------------------------
extern "C" void kernel_launch(void* const* d_in, const int* in_sizes, int n_in,
                              void* d_out, int out_size, void* d_ws,
                              size_t ws_size, hipStream_t stream) {
  (void)in_sizes; (void)n_in; (void)out_size; (void)ws_size;
  const float* x = (const float*)d_in[0];
  const float* W = (const float*)d_in[1];
  const float* b = (const float*)d_in[2];
  const float* S0 = (const float*)d_in[3];
  const float* Z0 = (const float*)d_in[4];
  float* out = (float*)d_out;

  float* Qw = (float*)d_ws;            // B*T*D
  float* Kw = Qw + (size_t)BT_ * D_;   // B*T*D
  float* Vw = Kw + (size_t)BT_ * D_;   // B*T*D

  qkv_proj_kernel<<<dim3(BT_ / 16), dim3(32), 0, stream>>>(x, W, b, Qw, Kw,
                                                           Vw);
  linattn_scan_kernel<<<dim3(NLBLK, B_), dim3(256),
                        SMEM_FLOATS * sizeof(float), stream>>>(Qw, Kw, Vw, S0,
                                                               Z0, out);
}